// MetaUpSample_9131100471706
// MI455X (gfx1250) — compile-verified
//
#include <hip/hip_runtime.h>

// MetaUpSample: out[b,ho,wo,f] = sum_k patch[b,ho,wo,k] * meta_w[b,ho,wo,k,f]
// B=4, H=W=64, C=64, Ho=Wo=128, K = 3*3*64 = 576, FILTERS = 3.
// Memory-bound on meta_w (453 MB streamed once); floor ~19.5us @ 23.3 TB/s.

typedef float v2f __attribute__((ext_vector_type(2)));
typedef float v8f __attribute__((ext_vector_type(8)));

// Exact fp32 sum of one value across all 32 lanes of a wave, computed on the
// WMMA pipe (V_WMMA_F32_16X16X4_F32). EXEC must be all ones here (it is: no
// divergence at the call sites).
//
// Step 1: A(vgpr0)=a, A(vgpr1)=0, B=ones:
//   A[m,K0]=a(lane m), A[m,K2]=a(lane m+16)  (documented 16x4 f32 A layout)
//   => D1[m,n] = a(m) + a(m+16) = s(m), replicated over n.
// D layout: VGPR r holds s(r) in lanes 0-15 and s(r+8) in lanes 16-31.
// Step 2 (x4, chained through C): A=(d1[2j], d1[2j+1]), B=ones adds
//   s(2j)+s(2j+1)+s(2j+8)+s(2j+9); after j=0..3 every lane holds sum s(0..15)
//   = sum of all 32 lanes.
__device__ __forceinline__ float wave32_sum_wmma(float a) {
  v2f A;    A.x = a;    A.y = 0.0f;
  v2f ones; ones.x = 1.0f; ones.y = 1.0f;
  v8f z = {};
  v8f d1 = __builtin_amdgcn_wmma_f32_16x16x4_f32(
      false, A, false, ones, (short)0, z, false, false);
  v8f acc = {};
#pragma unroll
  for (int j = 0; j < 4; ++j) {
    v2f Aj; Aj.x = d1[2 * j]; Aj.y = d1[2 * j + 1];
    acc = __builtin_amdgcn_wmma_f32_16x16x4_f32(
        false, Aj, false, ones, (short)0, acc, false, false);
  }
  return acc[0];
}

__global__ __launch_bounds__(256) void meta_upsample_kernel(
    const float* __restrict__ x,      // [4,64,64,64]
    const float* __restrict__ mw,     // [4,128,128,1728]  (k-major, f minor)
    float* __restrict__ out)          // [4,128,128,3]
{
  constexpr int H = 64, W = 64, C = 64, Ho = 128, Wo = 128;
  constexpr int KF = 576 * 3;  // 1728 floats per pixel

  // Shared x tile: 3 rows x 6 cols x 64 ch covering the 8 output pixels of
  // this block (wo = bx*8 .. bx*8+7  ->  ws in [bx*4, bx*4+3], +/-1 halo).
  __shared__ float tile[3 * 6 * 64];

  const int bx = blockIdx.x;        // Wo tile (16 tiles of 8)
  const int ho = blockIdx.y;        // 0..127
  const int b  = blockIdx.z;        // 0..3
  const int t  = threadIdx.x;       // 0..255 (8 wave32s)

  const int hs  = ho >> 1;          // source row  (floor(64*ho/128))
  const int ws0 = bx * 4;           // first source col used by this tile

  // Cooperative, coalesced tile load with zero padding (SAME padding).
  for (int idx = t; idx < 3 * 6 * 64; idx += 256) {
    const int dy  = idx / 384;          // 0..2
    const int rem = idx - dy * 384;
    const int tc  = rem >> 6;           // 0..5
    const int c   = rem & 63;
    const int sr  = hs - 1 + dy;
    const int sc  = ws0 - 1 + tc;
    float v = 0.0f;
    if (sr >= 0 && sr < H && sc >= 0 && sc < W)
      v = x[(((size_t)b * H + sr) * W + sc) * C + c];
    tile[idx] = v;
  }
  __syncthreads();

  const int lane = t & 31;
  const int wave = t >> 5;
  const int wo   = bx * 8 + wave;       // one wave per output pixel
  const int ws   = wo >> 1;
  const int colbase = ws - ws0;         // 0..3 within the 6-col tile (before +dx)

  const float* __restrict__ wp =
      mw + ((size_t)((b * Ho + ho) * Wo + wo)) * KF;

  float a0 = 0.0f, a1 = 0.0f, a2 = 0.0f;

  // k = 32*i + lane; lane loads 3 consecutive dwords w[k][0..2] -> the wave
  // covers a contiguous 384B span per iteration (fully coalesced), marked
  // non-temporal (meta_w is single-use; keep it out of L2's way).
#pragma unroll 6
  for (int i = 0; i < 18; ++i) {
    const int k   = i * 32 + lane;          // 0..575
    const int dy  = (unsigned)k / 192u;     // 0..2
    const int rem = k - dy * 192;
    const int dx  = rem >> 6;               // 0..2
    const int c   = rem & 63;
    // Bank-conflict-free: base is a multiple of 64 dwords, lanes span 32
    // consecutive c values mod 64.
    const float p = tile[(dy * 6 + colbase + dx) * 64 + c];

    const float* q = wp + k * 3;
    const float w0 = __builtin_nontemporal_load(q);
    const float w1 = __builtin_nontemporal_load(q + 1);
    const float w2 = __builtin_nontemporal_load(q + 2);

    a0 = fmaf(p, w0, a0);
    a1 = fmaf(p, w1, a1);
    a2 = fmaf(p, w2, a2);
  }

  // Cross-lane reductions on the matrix pipe (exact fp32).
  const float s0 = wave32_sum_wmma(a0);
  const float s1 = wave32_sum_wmma(a1);
  const float s2 = wave32_sum_wmma(a2);

  if (lane == 0) {
    const size_t o = (size_t)((b * Ho + ho) * Wo + wo) * 3;
    out[o + 0] = s0;
    out[o + 1] = s1;
    out[o + 2] = s2;
  }
}

extern "C" void kernel_launch(void* const* d_in, const int* in_sizes, int n_in,
                              void* d_out, int out_size, void* d_ws, size_t ws_size,
                              hipStream_t stream) {
  (void)in_sizes; (void)n_in; (void)out_size; (void)d_ws; (void)ws_size;
  const float* x  = (const float*)d_in[0];   // [4,64,64,64]
  const float* mw = (const float*)d_in[1];   // [4,128,128,1728]
  float* out = (float*)d_out;                // [4,128,128,3]

  dim3 grid(16, 128, 4);   // (Wo/8, Ho, B); 8 waves per block, 1 pixel/wave
  meta_upsample_kernel<<<grid, 256, 0, stream>>>(x, mw, out);
}